// MultiBoxLoss_89816356094037
// MI455X (gfx1250) — compile-verified
//
#include <hip/hip_runtime.h>
#include <hip/hip_bf16.h>

// ---------------- problem constants ----------------
constexpr int B_ = 128;
constexpr int N_ = 21824;          // divisible by 4 (b128 units) ; 8 slices of 2728
constexpr int SLICES = 8;
constexpr int PER = N_ / SLICES;   // 2728
constexpr int T1 = 256;
constexpr int T2 = 256;
constexpr int CHUNK = 4096;        // floats staged per chunk (16 KB)
constexpr int UNITS = CHUNK / 4;   // 1024 x b128
constexpr int IPT = UNITS / T2;    // 4 async issues per thread per chunk

// ---------------- CDNA5 async global->LDS helpers (guarded) ----------------
typedef int v4i_ __attribute__((ext_vector_type(4)));
typedef __attribute__((address_space(1))) v4i_ gv4i;   // global int32x4
typedef __attribute__((address_space(3))) v4i_ lv4i;   // LDS int32x4

#if __has_builtin(__builtin_amdgcn_global_load_async_to_lds_b128) && \
    __has_builtin(__builtin_amdgcn_s_wait_asynccnt)
#define HAS_ASYNC_LDS 1
#else
#define HAS_ASYNC_LDS 0
#endif

__device__ __forceinline__ void async_copy_b128(void* lds_dst, const void* gsrc) {
#if HAS_ASYNC_LDS
  __builtin_amdgcn_global_load_async_to_lds_b128((gv4i*)gsrc, (lv4i*)lds_dst, 0, 0);
#else
  *reinterpret_cast<float4*>(lds_dst) = *reinterpret_cast<const float4*>(gsrc);
#endif
}

template <int Nleft>
__device__ __forceinline__ void async_wait() {
#if HAS_ASYNC_LDS
  __builtin_amdgcn_s_wait_asynccnt(Nleft);
#endif
}

// ---------------- kernel 1: fused ce/smooth-L1/mining streaming pass ----------------
__global__ __launch_bounds__(T1) void mbox_stage1(
    const float* __restrict__ loc_p, const float* __restrict__ loc_t,
    const float* __restrict__ conf,  const int*  __restrict__ tgt,
    float* __restrict__ mining, float* __restrict__ part_loc,
    float* __restrict__ part_pce, int* __restrict__ part_np)
{
  const int blk = blockIdx.x;
  const int b = blk / SLICES;
  const int s = blk % SLICES;
  const int tid = threadIdx.x;
  const long base = (long)b * N_;

  float lsum = 0.f, psum = 0.f;
  int np = 0;

  const int n_end = (s + 1) * PER;
  for (int n = s * PER + tid; n < n_end; n += T1) {
    const long i = base + n;
    // prefetch next iteration's loc streams (lowers to global_prefetch_b8)
    __builtin_prefetch(loc_p + 4 * (i + T1), 0, 0);
    __builtin_prefetch(loc_t + 4 * (i + T1), 0, 0);

    const int t = tgt[i];
    const bool pos = t > 0;

    const float2 c = *reinterpret_cast<const float2*>(conf + 2 * i);
    const float m  = fmaxf(c.x, c.y);
    const float lse = m + __logf(__expf(c.x - m) + __expf(c.y - m));
    const float ce  = lse - (t ? c.y : c.x);

    mining[i] = pos ? 0.f : fmaxf(ce, 0.f);   // non-negative -> u32 bit order == float order

    const float4 p = *reinterpret_cast<const float4*>(loc_p + 4 * i);
    const float4 q = *reinterpret_cast<const float4*>(loc_t + 4 * i);
    float sl = 0.f;
    { float d = p.x - q.x, a = fabsf(d); sl += (a < 1.f) ? 0.5f * d * d : a - 0.5f; }
    { float d = p.y - q.y, a = fabsf(d); sl += (a < 1.f) ? 0.5f * d * d : a - 0.5f; }
    { float d = p.z - q.z, a = fabsf(d); sl += (a < 1.f) ? 0.5f * d * d : a - 0.5f; }
    { float d = p.w - q.w, a = fabsf(d); sl += (a < 1.f) ? 0.5f * d * d : a - 0.5f; }

    if (pos) { lsum += sl; psum += ce; np += 1; }
  }

  __shared__ float sf[T1];
  __shared__ int   si[T1];

  sf[tid] = lsum; __syncthreads();
  for (int o = T1 / 2; o > 0; o >>= 1) { if (tid < o) sf[tid] += sf[tid + o]; __syncthreads(); }
  if (tid == 0) part_loc[blk] = sf[0];
  __syncthreads();

  sf[tid] = psum; __syncthreads();
  for (int o = T1 / 2; o > 0; o >>= 1) { if (tid < o) sf[tid] += sf[tid + o]; __syncthreads(); }
  if (tid == 0) part_pce[blk] = sf[0];
  __syncthreads();

  si[tid] = np; __syncthreads();
  for (int o = T1 / 2; o > 0; o >>= 1) { if (tid < o) si[tid] += si[tid + o]; __syncthreads(); }
  if (tid == 0) part_np[blk] = si[0];
}

// ---------------- kernel 2: per-row radix top-k select (async-staged) ----------------
__global__ __launch_bounds__(T2) void mbox_select(
    const float* __restrict__ mining, const int* __restrict__ part_np,
    float* __restrict__ row_neg)
{
  const int b   = blockIdx.x;
  const int tid = threadIdx.x;

  __shared__ __align__(16) float stage[2][CHUNK];
  __shared__ int   hist[2048];
  __shared__ int   s_need, s_bucket;
  __shared__ float s_red[T2];

  if (tid == 0) {
    int np = 0;
    for (int j = 0; j < SLICES; ++j) np += part_np[b * SLICES + j];
    long k = 3L * np;
    if (k > (long)(N_ - 1)) k = N_ - 1;
    s_need = (int)k;
  }
  __syncthreads();
  const int k0 = s_need;
  if (k0 == 0) { if (tid == 0) row_neg[b] = 0.f; return; }

  const float* row = mining + (long)b * N_;

  auto issue = [&](int c, int buf) {
    const int start = c * CHUNK;
    int remf = N_ - start; if (remf > CHUNK) remf = CHUNK;
    const int remu = remf >> 2;                 // N_ % 4 == 0
#pragma unroll
    for (int j = 0; j < IPT; ++j) {
      int u = j * T2 + tid;
      int uu = (u < remu) ? u : (remu - 1);     // clamp: all threads issue IPT (const wait)
      async_copy_b128(&stage[buf][uu * 4], row + start + uu * 4);
    }
  };

  auto stream = [&](auto fn) {
    const int nC = (N_ + CHUNK - 1) / CHUNK;
    issue(0, 0);
    for (int c = 0; c < nC; ++c) {
      if (c + 1 < nC) { issue(c + 1, (c + 1) & 1); async_wait<IPT>(); }
      else            { async_wait<0>(); }
      __syncthreads();                           // chunk c visible to all waves
      const float* buf = stage[c & 1];
      int cnt = N_ - c * CHUNK; if (cnt > CHUNK) cnt = CHUNK;
      for (int i = tid; i < cnt; i += T2) fn(buf[i]);
      __syncthreads();                           // done reading before buffer reuse
    }
  };

  auto zero_hist = [&]() {
    for (int i = tid; i < 2048; i += T2) hist[i] = 0;
    __syncthreads();
  };

  auto scan_top = [&](int nb) {                  // consumes hist; updates s_need, s_bucket
    if (tid == 0) {
      int need = s_need, acc = 0, sel = nb - 1;
      for (int i = nb - 1; i >= 0; --i) {
        int h = hist[i];
        if (acc + h >= need) { sel = i; break; }
        acc += h;
      }
      s_bucket = sel;
      s_need   = need - acc;                     // rank within chosen bucket (>=1)
    }
    __syncthreads();
  };

  // level 1: bits [31:21]
  zero_hist();
  stream([&](float v) { unsigned u = __float_as_uint(v); atomicAdd(&hist[u >> 21], 1); });
  scan_top(2048);
  const unsigned p1 = (unsigned)s_bucket;

  // level 2: bits [20:10]
  zero_hist();
  stream([&](float v) {
    unsigned u = __float_as_uint(v);
    if ((u >> 21) == p1) atomicAdd(&hist[(u >> 10) & 0x7FF], 1);
  });
  scan_top(2048);
  const unsigned p2 = (p1 << 11) | (unsigned)s_bucket;

  // level 3: bits [9:0]
  zero_hist();
  stream([&](float v) {
    unsigned u = __float_as_uint(v);
    if ((u >> 10) == p2) atomicAdd(&hist[u & 0x3FF], 1);
  });
  scan_top(1024);
  const unsigned tbits = (p2 << 10) | (unsigned)s_bucket;
  const int ties = s_need;                       // slots filled by value == t
  __syncthreads();

  // final pass: sum of strictly-greater values
  float acc = 0.f;
  stream([&](float v) { if (__float_as_uint(v) > tbits) acc += v; });
  s_red[tid] = acc; __syncthreads();
  for (int o = T2 / 2; o > 0; o >>= 1) { if (tid < o) s_red[tid] += s_red[tid + o]; __syncthreads(); }
  if (tid == 0) row_neg[b] = s_red[0] + (float)ties * __uint_as_float(tbits);
}

// ---------------- kernel 3: deterministic finalize ----------------
__global__ __launch_bounds__(256) void mbox_final(
    const float* __restrict__ part_loc, const float* __restrict__ part_pce,
    const int* __restrict__ part_np, const float* __restrict__ row_neg,
    float* __restrict__ out)
{
  __shared__ float sf[256];
  __shared__ int   si[256];
  const int tid = threadIdx.x;
  float acc = 0.f;
  int nm = 0;
  for (int i = tid; i < B_ * SLICES; i += 256) {
    acc += part_loc[i] + part_pce[i];
    nm  += part_np[i];
  }
  for (int i = tid; i < B_; i += 256) acc += row_neg[i];
  sf[tid] = acc; si[tid] = nm; __syncthreads();
  for (int o = 128; o > 0; o >>= 1) {
    if (tid < o) { sf[tid] += sf[tid + o]; si[tid] += si[tid + o]; }
    __syncthreads();
  }
  if (tid == 0) {
    int m = si[0];
    out[0] = (m == 0) ? 0.f : sf[0] / fmaxf((float)m, 1.f);
  }
}

// ---------------- host-side launch ----------------
extern "C" void kernel_launch(void* const* d_in, const int* in_sizes, int n_in,
                              void* d_out, int out_size, void* d_ws, size_t ws_size,
                              hipStream_t stream)
{
  const float* loc_p = (const float*)d_in[0];
  const float* loc_t = (const float*)d_in[1];
  const float* conf  = (const float*)d_in[2];
  const int*   tgt   = (const int*)d_in[3];
  float* out = (float*)d_out;

  char* ws = (char*)d_ws;
  float* mining   = (float*)ws;  ws += sizeof(float) * (size_t)B_ * N_;
  float* part_loc = (float*)ws;  ws += sizeof(float) * B_ * SLICES;
  float* part_pce = (float*)ws;  ws += sizeof(float) * B_ * SLICES;
  int*   part_np  = (int*)ws;    ws += sizeof(int)   * B_ * SLICES;
  float* row_neg  = (float*)ws;

  mbox_stage1<<<B_ * SLICES, T1, 0, stream>>>(loc_p, loc_t, conf, tgt,
                                              mining, part_loc, part_pce, part_np);
  mbox_select<<<B_, T2, 0, stream>>>(mining, part_np, row_neg);
  mbox_final<<<1, 256, 0, stream>>>(part_loc, part_pce, part_np, row_neg, out);
}